// DennisNode_5068061409913
// MI455X (gfx1250) — compile-verified
//
#include <hip/hip_runtime.h>
#include <stdint.h>
#include <math.h>

#ifndef __has_builtin
#define __has_builtin(x) 0
#endif

// Problem constants (reference: T=1024, N=65536, THRESHOLD=0.01)
#define TT     1024
#define NN     65536
#define BLOCK  256
#define VEC    4
#define COLS   (BLOCK * VEC)   // 1024 columns per block
#define CHUNK  4               // time steps staged per LDS buffer
#define NBUF   4               // ring buffers (prefetch depth 3)
#define NCHUNK (TT / CHUNK)    // 256

#define AS1 __attribute__((address_space(1)))
#define AS3 __attribute__((address_space(3)))

typedef int   v4i __attribute__((ext_vector_type(4)));
typedef float v4f __attribute__((ext_vector_type(4)));

#if defined(__HIP_DEVICE_COMPILE__)

#if __has_builtin(__builtin_amdgcn_global_load_async_to_lds_b128)
#define HAS_ASYNC_LDS 1
#else
#define HAS_ASYNC_LDS 0
#endif

#if __has_builtin(__builtin_amdgcn_s_wait_asynccnt)
#define WAIT_ASYNC(n) do { __builtin_amdgcn_s_wait_asynccnt(n); asm volatile("" ::: "memory"); } while (0)
#else
#define WAIT_ASYNC(n) asm volatile("s_wait_asynccnt %0" :: "i"(n) : "memory")
#endif

#if __has_builtin(__builtin_amdgcn_s_wait_dscnt)
#define WAIT_DS(n) do { __builtin_amdgcn_s_wait_dscnt(n); asm volatile("" ::: "memory"); } while (0)
#else
#define WAIT_DS(n) asm volatile("s_wait_dscnt %0" :: "i"(n) : "memory")
#endif

#else  // host pass: never executed, keep it parseable
#define HAS_ASYNC_LDS 0
#define WAIT_ASYNC(n)
#define WAIT_DS(n)
#endif

// One recurrence step for 4 packed columns. Only (X1, Z, E) are live state:
// phase/health in the reference are dead w.r.t. the Z_all output.
__device__ __forceinline__ v4f step4(v4f z4, v4f n4,
                                     float* X1, float* Z, float* E,
                                     float cb, float ifw, float irv,
                                     float cp, float dp, float na) {
    v4f rs;
#pragma unroll
    for (int v = 0; v < 4; ++v) {
        float tn = fabsf(z4[v] - Z[v]);
        float e  = E[v] * 0.98f + tn * 100.0f;
        e = fminf(fmaxf(e, 0.0f), 1.0e6f);
        E[v] = e;
        float coup = (tn < 0.01f) ? -cb : cb;           // THRESHOLD = 0.01
        float x1 = X1[v] + ifw + e * 0.005f;
        x1 = x1 - cp * x1;
        X1[v] = x1;
        float x2  = x1 - irv;
        float phi = 0.5f * (x1 + x2);
        float x3  = phi * (1.0f - dp);
        float y   = fabsf(x3 - phi);
        float raw = x3 * y + coup * 0.1f + na * n4[v];
        float zz  = __builtin_isfinite(raw) ? raw : 0.0f;
        Z[v]  = zz;
        rs[v] = zz;
    }
    return rs;
}

__global__ __launch_bounds__(BLOCK) void dennis_scan(
    const float* __restrict__ zn, const float* __restrict__ nz,
    const float* __restrict__ p_cb, const float* __restrict__ p_if,
    const float* __restrict__ p_ir, const float* __restrict__ p_cp,
    const float* __restrict__ p_dp, const float* __restrict__ p_na,
    float* __restrict__ out) {
    const int tid = threadIdx.x;
    const long long col0 = (long long)blockIdx.x * COLS + (long long)tid * VEC;

    const float cb  = *p_cb;
    const float ifw = *p_if;
    const float irv = *p_ir;
    const float cp  = *p_cp;
    const float dp  = *p_dp;
    const float na  = *p_na;

    float X1[VEC], Z[VEC], E[VEC];
#pragma unroll
    for (int v = 0; v < VEC; ++v) { X1[v] = 0.0f; Z[v] = 0.0f; E[v] = 0.01f; }

#if HAS_ASYNC_LDS
    // 4-buffer ring, depth-3 async prefetch: NBUF x CHUNK rows x 1024 cols x
    // (zn,nz) = 128 KB LDS (<= 320 KB/WGP, 2 workgroups/WGP). In flight across
    // the device: 3 chunks x 65536 cols x 8 B = 6 MB. Each lane DMAs and reads
    // back only its own 16B slots -> no barrier, only per-wave ASYNCcnt/DScnt.
    __shared__ alignas(16) float sZ[NBUF][CHUNK][COLS];
    __shared__ alignas(16) float sN[NBUF][CHUNK][COLS];

    const float* zt = zn + col0;
    const float* nt = nz + col0;

#define ISSUE(c)                                                              \
    do {                                                                      \
        const int _s = (c) & (NBUF - 1);                                      \
        const float* _zb = zt + (long long)(c) * CHUNK * NN;                  \
        const float* _nb = nt + (long long)(c) * CHUNK * NN;                  \
        _Pragma("unroll")                                                     \
        for (int _r = 0; _r < CHUNK; ++_r) {                                  \
            __builtin_amdgcn_global_load_async_to_lds_b128(                   \
                (AS1 v4i*)(_zb + (long long)_r * NN),                         \
                (AS3 v4i*)&sZ[_s][_r][tid * VEC], 0, 0);                      \
            __builtin_amdgcn_global_load_async_to_lds_b128(                   \
                (AS1 v4i*)(_nb + (long long)_r * NN),                         \
                (AS3 v4i*)&sN[_s][_r][tid * VEC], 0, 0);                      \
        }                                                                     \
    } while (0)

    ISSUE(0);
    ISSUE(1);
    ISSUE(2);
    for (int c = 0; c < NCHUNK; ++c) {
        // Async loads complete in order; 8 ops per chunk. Allow the chunks
        // still wanted in flight (c+1, c+2) to remain outstanding while
        // guaranteeing chunk c has landed in LDS.
        if (c + 2 < NCHUNK)      { WAIT_ASYNC(16); }
        else if (c + 1 < NCHUNK) { WAIT_ASYNC(8); }
        else                     { WAIT_ASYNC(0); }

        const int s = c & (NBUF - 1);
        float* po = out + (long long)c * CHUNK * NN + col0;
#pragma unroll
        for (int r = 0; r < CHUNK; ++r) {
            v4f z4 = *(const v4f*)&sZ[s][r][tid * VEC];
            v4f n4 = *(const v4f*)&sN[s][r][tid * VEC];
            v4f o4 = step4(z4, n4, X1, Z, E, cb, ifw, irv, cp, dp, na);
            // Write-once 256 MB output stream: non-temporal, keep L2 for inputs.
            __builtin_nontemporal_store(o4, (v4f*)(po + (long long)r * NN));
        }
        WAIT_DS(0);  // LDS reads retired before the DMA engine reuses a buffer
        if (c + 3 < NCHUNK) ISSUE(c + 3);
    }
#undef ISSUE
#else
    // Fallback: register double-buffered streaming loads.
    v4f z4 = *(const v4f*)(zn + col0);
    v4f n4 = *(const v4f*)(nz + col0);
    for (int t = 0; t < TT; ++t) {
        v4f z4n = z4, n4n = n4;
        if (t + 1 < TT) {
            z4n = *(const v4f*)(zn + (long long)(t + 1) * NN + col0);
            n4n = *(const v4f*)(nz + (long long)(t + 1) * NN + col0);
        }
        v4f o4 = step4(z4, n4, X1, Z, E, cb, ifw, irv, cp, dp, na);
        __builtin_nontemporal_store(o4, (v4f*)(out + (long long)t * NN + col0));
        z4 = z4n; n4 = n4n;
    }
#endif
}

extern "C" void kernel_launch(void* const* d_in, const int* in_sizes, int n_in,
                              void* d_out, int out_size, void* d_ws, size_t ws_size,
                              hipStream_t stream) {
    (void)in_sizes; (void)n_in; (void)out_size; (void)d_ws; (void)ws_size;
    const float* zn = (const float*)d_in[0];
    const float* nz = (const float*)d_in[1];
    dennis_scan<<<NN / COLS, BLOCK, 0, stream>>>(
        zn, nz,
        (const float*)d_in[2], (const float*)d_in[3], (const float*)d_in[4],
        (const float*)d_in[5], (const float*)d_in[6], (const float*)d_in[7],
        (float*)d_out);
}